// OpBPM_par_37151467111013
// MI455X (gfx1250) — compile-verified
//
#include <hip/hip_runtime.h>
#include <math.h>

// ---------------------------------------------------------------------------
// Split-step Fourier BPM on gfx1250.
// One workgroup (256 threads = 8 wave32) per (batch, angle) row.
// FFT2048 = four-step: DFT16 via v_wmma_f32_16x16x4_f32 (complex as 4 real
// matmuls; +/-Wi sign baked into the A operand since f32 WMMA has no A-neg
// modifier) + 7 radix-2 LDS stages for the 128-point part.
// Frequency-domain multiplies stay in the permuted (bit-reversed) domain.
// ---------------------------------------------------------------------------

typedef float v2f __attribute__((ext_vector_type(2)));
typedef float v8f __attribute__((ext_vector_type(8)));

#define NXE   2048     // extended line length
#define N2    128      // 2048 = 16 x 128
#define NXC   512      // central (unpadded) width
#define NZC   128      // z slices
#define THREADS 256

__device__ __forceinline__ int rev7(int v) { return __brev(v) >> 25; }

struct Smem {
  float ur[NXE], ui[NXE];   // field (real / imag)
  float wr[NXE], wi[NXE];   // w[k] = exp(-2*pi*i*k/2048)
  float fr[NXE], fi[NXE];   // permuted+scaled frequency filter (f1 or f2)
  float xs[NXC];            // current x slice
};

// ---- radix-16 stage as WMMA matmul: Y = W16 * X  (X viewed as 16x128) ------
// Forward: W16[k1][n1] = w[(k1*n1 mod 16)*128]; fused twiddle w[k1*n2] on store.
// Inverse: A = conj(W16) (sign folded into Wi operand), fused conj-twiddle on
// B load.  Complex matmul:
//   fwd: Cr = Wr*Br - Wi*Bi ; Ci = Wr*Bi + Wi*Br
//   inv: Cr = Wr*Br + Wi*Bi ; Ci = Wr*Bi - Wi*Br
template<bool INV>
__device__ __forceinline__ void dft16_wmma(Smem& s, int lane, int wave) {
  const int Mhalf = lane >> 4;          // 0 or 1
  const int Ml    = lane & 15;
  const int col   = (wave << 4) + Ml;   // this wave's n2 column

  v2f ar[4], aip[4], ain[4], br[4], bi[4];
  #pragma unroll
  for (int kc = 0; kc < 4; ++kc) {
    const int kk = 4 * kc + 2 * Mhalf;          // K index for .x; .y is kk+1
    // A operand: rows of W16 (M = Ml, K = kk/kk+1)
    const int e0 = ((Ml * kk) & 15) << 7;
    const int e1 = ((Ml * (kk + 1)) & 15) << 7;
    v2f a0; a0.x = s.wr[e0]; a0.y = s.wr[e1]; ar[kc] = a0;
    v2f a1; a1.x = s.wi[e0]; a1.y = s.wi[e1];
    aip[kc] = a1;            // +Wi
    ain[kc] = -a1;           // -Wi  (f32 WMMA has no A-neg modifier)
    // B operand: X rows K = kk/kk+1, column = col
    const float yr0 = s.ur[kk * N2 + col],       yi0 = s.ui[kk * N2 + col];
    const float yr1 = s.ur[(kk + 1) * N2 + col], yi1 = s.ui[(kk + 1) * N2 + col];
    v2f b0, b1;
    if (INV) {
      // multiply by conj(w[k1*n2]) while loading  (k1 = kk, n2 = col)
      const int eb0 = kk * col, eb1 = (kk + 1) * col;   // < 2048
      const float c0 = s.wr[eb0], s0 = s.wi[eb0];
      const float c1 = s.wr[eb1], s1 = s.wi[eb1];
      b0.x = yr0 * c0 + yi0 * s0;  b0.y = yr1 * c1 + yi1 * s1;
      b1.x = yi0 * c0 - yr0 * s0;  b1.y = yi1 * c1 - yr1 * s1;
    } else {
      b0.x = yr0; b0.y = yr1;
      b1.x = yi0; b1.y = yi1;
    }
    br[kc] = b0; bi[kc] = b1;
  }

  v8f cr = {}, ci = {};
  #pragma unroll
  for (int kc = 0; kc < 4; ++kc) {
    cr = __builtin_amdgcn_wmma_f32_16x16x4_f32(false, ar[kc], false, br[kc],
                                               (short)0, cr, false, false);
    ci = __builtin_amdgcn_wmma_f32_16x16x4_f32(false, ar[kc], false, bi[kc],
                                               (short)0, ci, false, false);
  }
  #pragma unroll
  for (int kc = 0; kc < 4; ++kc) {
    // fwd: Cr += (-Wi)*Bi, Ci += (+Wi)*Br ; inv: Cr += (+Wi)*Bi, Ci += (-Wi)*Br
    cr = __builtin_amdgcn_wmma_f32_16x16x4_f32(false, INV ? aip[kc] : ain[kc],
                                               false, bi[kc],
                                               (short)0, cr, false, false);
    ci = __builtin_amdgcn_wmma_f32_16x16x4_f32(false, INV ? ain[kc] : aip[kc],
                                               false, br[kc],
                                               (short)0, ci, false, false);
  }

  #pragma unroll
  for (int r = 0; r < 8; ++r) {
    const int M = r + (Mhalf << 3);      // output row (k1 fwd, n1 inv)
    float orr = cr[r], oii = ci[r];
    if (!INV) {                          // fused twiddle w[k1*n2] on store
      const int e = M * col;             // < 2048
      const float c = s.wr[e], sn = s.wi[e];
      const float tr2 = orr * c - oii * sn;
      const float ti2 = orr * sn + oii * c;
      orr = tr2; oii = ti2;
    }
    s.ur[M * N2 + col] = orr;
    s.ui[M * N2 + col] = oii;
  }
}

// ---- one radix-2 stage of the 16 parallel 128-point FFTs -------------------
// Forward (DIF): natural->bitrev.  Inverse (DIT): bitrev->natural, conj tw.
// First inverse stage (shift==0, Fr!=null) fuses the frequency filter (tw==1).
template<bool INV>
__device__ __forceinline__ void fft128_stage(Smem& s, int tid, int shift,
                                             const float* Fr, const float* Fi) {
  const int half = 1 << shift;
  #pragma unroll
  for (int it = 0; it < 4; ++it) {
    const int t   = tid + it * THREADS;      // 1024 butterflies total
    const int row = t >> 6;
    const int q   = t & 63;
    const int j   = q & (half - 1);
    const int g   = q >> shift;
    const int i1  = (row << 7) + (g << (shift + 1)) + j;
    const int i2  = i1 + half;
    const int jt  = j << (10 - shift);       // twiddle index into w2048
    float ar_ = s.ur[i1], ai_ = s.ui[i1];
    float br_ = s.ur[i2], bi_ = s.ui[i2];
    const float c = s.wr[jt], sn = s.wi[jt];
    if (!INV) {
      s.ur[i1] = ar_ + br_;  s.ui[i1] = ai_ + bi_;
      const float dr = ar_ - br_, di = ai_ - bi_;
      s.ur[i2] = dr * c - di * sn;
      s.ui[i2] = dr * sn + di * c;
    } else {
      if (Fr) {  // shift==0: tw==1, apply filter to both inputs
        const float f0r = Fr[i1], f0i = Fi[i1];
        const float f1r = Fr[i2], f1i = Fi[i2];
        const float t0r = ar_ * f0r - ai_ * f0i, t0i = ar_ * f0i + ai_ * f0r;
        const float t1r = br_ * f1r - bi_ * f1i, t1i = br_ * f1i + bi_ * f1r;
        ar_ = t0r; ai_ = t0i; br_ = t1r; bi_ = t1i;
      } else {   // conj twiddle on b
        const float tr2 = br_ * c + bi_ * sn;
        const float ti2 = bi_ * c - br_ * sn;
        br_ = tr2; bi_ = ti2;
      }
      s.ur[i1] = ar_ + br_;  s.ui[i1] = ai_ + bi_;
      s.ur[i2] = ar_ - br_;  s.ui[i2] = ai_ - bi_;
    }
  }
}

// ---- u <- IFFT( FFT(u) * F )  (F pre-permuted, pre-scaled by 1/2048) -------
__device__ __forceinline__ void roundtrip(Smem& s, int tid, int lane, int wave,
                                          const float* Fr, const float* Fi) {
  __syncthreads();
  dft16_wmma<false>(s, lane, wave);
  __syncthreads();
  #pragma unroll
  for (int shift = 6; shift >= 0; --shift) {
    fft128_stage<false>(s, tid, shift, nullptr, nullptr);
    __syncthreads();
  }
  fft128_stage<true>(s, tid, 0, Fr, Fi);   // fused filter multiply
  __syncthreads();
  #pragma unroll
  for (int shift = 1; shift <= 6; ++shift) {
    fft128_stage<true>(s, tid, shift, nullptr, nullptr);
    __syncthreads();
  }
  dft16_wmma<true>(s, lane, wave);
  __syncthreads();
}

__global__ __launch_bounds__(THREADS)
void bpm_kernel(const float* __restrict__ x,  const float* __restrict__ uin,
                const float* __restrict__ f1, const float* __restrict__ f2,
                const float* __restrict__ costh, const float* __restrict__ k0,
                const float* __restrict__ drz,   const int* __restrict__ pad,
                float* __restrict__ out, int NUM)
{
  __shared__ Smem s;
  const int tid  = threadIdx.x;
  const int lane = tid & 31;
  const int wave = tid >> 5;
  const int blk  = blockIdx.x;
  const int n    = blk % NUM;
  const int b    = blk / NUM;
  const float alpha = k0[0] * drz[0] / costh[n];
  const int   p0    = pad[0];

  // init: twiddle table, permuted/scaled f1, broadcast uin
  const float thstep = -6.283185307179586f / (float)NXE;
  for (int i = tid; i < NXE; i += THREADS) {
    const float th = thstep * (float)i;
    s.wr[i] = cosf(th);
    s.wi[i] = sinf(th);
    const int k = (i >> 7) + (rev7(i & 127) << 4);   // freq at permuted slot i
    s.fr[i] = f1[2 * k]     * (1.0f / NXE);
    s.fi[i] = f1[2 * k + 1] * (1.0f / NXE);
    s.ur[i] = uin[2 * i];
    s.ui[i] = uin[2 * i + 1];
  }

  const float* xb = x + (size_t)b * (NXC * NZC);
  for (int z = 0; z < NZC; ++z) {
    // load this z slice (stride NZC gather; lines stay hot across z)
    for (int i = tid; i < NXC; i += THREADS) {
      s.xs[i] = xb[i * NZC + z];
      __builtin_prefetch(&xb[i * NZC + ((z + 16) & (NZC - 1))], 0, 1);
    }
    roundtrip(s, tid, lane, wave, s.fr, s.fi);   // leading barrier covers xs
    // central phase screen: u *= exp(i * alpha * x)
    for (int i = tid; i < NXC; i += THREADS) {
      const int idx = p0 + i;
      float sn, cs;
      __sincosf(alpha * s.xs[i], &sn, &cs);
      const float urv = s.ur[idx], uiv = s.ui[idx];
      s.ur[idx] = urv * cs - uiv * sn;
      s.ui[idx] = urv * sn + uiv * cs;
    }
    __syncthreads();
  }

  // propagate to sensor: rebuild filter from f2, final roundtrip
  for (int i = tid; i < NXE; i += THREADS) {
    const int k = (i >> 7) + (rev7(i & 127) << 4);
    s.fr[i] = f2[2 * k]     * (1.0f / NXE);
    s.fi[i] = f2[2 * k + 1] * (1.0f / NXE);
  }
  roundtrip(s, tid, lane, wave, s.fr, s.fi);

  float* o = out + (size_t)blk * NXE * 2;
  for (int i = tid; i < NXE; i += THREADS) {
    o[2 * i]     = s.ur[i];
    o[2 * i + 1] = s.ui[i];
  }
}

extern "C" void kernel_launch(void* const* d_in, const int* in_sizes, int n_in,
                              void* d_out, int out_size, void* d_ws, size_t ws_size,
                              hipStream_t stream) {
  const float* x     = (const float*)d_in[0];
  const float* uin   = (const float*)d_in[1];
  const float* f1    = (const float*)d_in[2];
  const float* f2    = (const float*)d_in[3];
  const float* costh = (const float*)d_in[4];
  const float* k0    = (const float*)d_in[5];
  const float* drz   = (const float*)d_in[6];
  const int*   pad   = (const int*)d_in[7];

  const int NUM = in_sizes[4];                 // 48
  const int NB  = in_sizes[0] / (NXC * NZC);   // 2
  dim3 grid(NB * NUM), block(THREADS);
  hipLaunchKernelGGL(bpm_kernel, grid, block, 0, stream,
                     x, uin, f1, f2, costh, k0, drz, pad, (float*)d_out, NUM);
}